// EncoderDecoder_35966056136758
// MI455X (gfx1250) — compile-verified
//
#include <hip/hip_runtime.h>
#include <hip/hip_bf16.h>
#include <math.h>

typedef __attribute__((ext_vector_type(2))) float v2f;
typedef __attribute__((ext_vector_type(8))) float v8f;

#define Hd 2048
#define Id 512
#define Od 512
#define G3 6144      // 3*H
#define NBLK 64
#define NTHR 256

// ---------------------------------------------------------------------------
// init: zero ping-pong hidden state, barrier counters, feedback index
// ---------------------------------------------------------------------------
__global__ void gru_init_kernel(float* h, unsigned* cnt, unsigned* gen, int* idxp) {
    int t = threadIdx.x + blockIdx.x * blockDim.x;
    for (int i = t; i < 2 * Hd; i += blockDim.x * gridDim.x) h[i] = 0.0f;
    if (t == 0) { *cnt = 0u; *gen = 0u; *idxp = -1; }
}

// ---------------------------------------------------------------------------
// WMMA GEMM:  Gi(L,6144) = X(L,512) * Wih^T(512,6144) + b_ih
// fp32 WMMA 16x16x4. Register-blocked: one wave owns a 16x64 strip
// (4 accumulators), so each A fragment is reused by 4 WMMAs and the four
// WMMA chains are independent (hides the D->A/B RAW hazard window).
// A frag (16x4 f32): lane<16 holds {K,K+1} of row M=lane&15, lane>=16 {K+2,K+3}.
// B frag mirrors with N<->M; B[k][n] = Wih[n][k] so all loads are contiguous b64.
// ---------------------------------------------------------------------------
__global__ void __launch_bounds__(NTHR)
gi_gemm_kernel(const float* __restrict__ X, const float* __restrict__ W,
               const float* __restrict__ bias, float* __restrict__ Gi, int L) {
    const int lane = threadIdx.x & 31;
    const int wid  = blockIdx.x * (NTHR >> 5) + (threadIdx.x >> 5);
    const int strips_n = G3 / 64;                // 96 strips of 64 columns
    const int tm = wid / strips_n;
    const int sn = wid - tm * strips_n;
    if (tm * 16 >= L) return;                    // wave-uniform: EXEC stays all-1

    const int hi  = (lane >= 16) ? 1 : 0;
    const int l15 = lane & 15;
    const int m  = tm * 16 + l15;                // A row held by this lane
    const int n0 = sn * 64 + l15;                // first B/D column of the strip

    const float* arow = X + (size_t)m * Id + hi * 2;
    const float* b0 = W + (size_t)(n0     ) * Id + hi * 2;
    const float* b1 = W + (size_t)(n0 + 16) * Id + hi * 2;
    const float* b2 = W + (size_t)(n0 + 32) * Id + hi * 2;
    const float* b3 = W + (size_t)(n0 + 48) * Id + hi * 2;

    v8f acc0 = {}, acc1 = {}, acc2 = {}, acc3 = {};
    #pragma unroll 4
    for (int k = 0; k < Id; k += 4) {
        v2f a  = *(const v2f*)(arow + k);
        v2f f0 = *(const v2f*)(b0 + k);
        v2f f1 = *(const v2f*)(b1 + k);
        v2f f2 = *(const v2f*)(b2 + k);
        v2f f3 = *(const v2f*)(b3 + k);
        acc0 = __builtin_amdgcn_wmma_f32_16x16x4_f32(false, a, false, f0,
                                                     (short)0, acc0, false, false);
        acc1 = __builtin_amdgcn_wmma_f32_16x16x4_f32(false, a, false, f1,
                                                     (short)0, acc1, false, false);
        acc2 = __builtin_amdgcn_wmma_f32_16x16x4_f32(false, a, false, f2,
                                                     (short)0, acc2, false, false);
        acc3 = __builtin_amdgcn_wmma_f32_16x16x4_f32(false, a, false, f3,
                                                     (short)0, acc3, false, false);
    }
    const float bv0 = bias[n0], bv1 = bias[n0 + 16];
    const float bv2 = bias[n0 + 32], bv3 = bias[n0 + 48];
    float* dbase = Gi + (size_t)(tm * 16 + hi * 8) * G3;     // D vgpr g -> row g+hi*8
    #pragma unroll
    for (int g = 0; g < 8; ++g) {
        float* dr = dbase + (size_t)g * G3;
        dr[n0]      = acc0[g] + bv0;
        dr[n0 + 16] = acc1[g] + bv1;
        dr[n0 + 32] = acc2[g] + bv2;
        dr[n0 + 48] = acc3[g] + bv3;
    }
}

// ---------------------------------------------------------------------------
// persistent recurrent kernel: encoder scan + autoregressive decoder
// ---------------------------------------------------------------------------
__device__ __forceinline__ void grid_barrier(unsigned* cnt, unsigned* gen) {
    __syncthreads();
    if (threadIdx.x == 0) {
        __threadfence();
        unsigned g = atomicAdd(gen, 0u);
        if (atomicAdd(cnt, 1u) == NBLK - 1) {
            atomicExch(cnt, 0u);
            atomicAdd(gen, 1u);
        } else {
            while (atomicAdd(gen, 0u) == g) __builtin_amdgcn_s_sleep(2);
        }
        __threadfence();
    }
    __syncthreads();
}

__device__ __forceinline__ float wave_sum(float v) {
    #pragma unroll
    for (int o = 16; o > 0; o >>= 1) v += __shfl_xor(v, o, 32);
    return v;
}
__device__ __forceinline__ float sigm(float x) { return 1.0f / (1.0f + __expf(-x)); }

__global__ void __launch_bounds__(NTHR, 1)
rnn_kernel(const float* __restrict__ Gi,
           const float* __restrict__ eWhh, const float* __restrict__ eBhh,
           const float* __restrict__ dWih, const float* __restrict__ dWhh,
           const float* __restrict__ dBih, const float* __restrict__ dBhh,
           const float* __restrict__ h2oW, const float* __restrict__ h2oB,
           float* h, float* logits, unsigned* cnt, unsigned* gen, int* idxp,
           float* __restrict__ out, int L, int T) {
    __shared__ float hS[Hd];
    __shared__ float red[NTHR];
    __shared__ int   redi[NTHR];
    const int tid  = threadIdx.x;
    const int lane = tid & 31;
    const int wv   = tid >> 5;      // 0..7
    const int blk  = blockIdx.x;
    int p = 0;

    // ---------------- encoder: L sequential steps ----------------
    for (int t = 0; t < L; ++t) {
        const float* hc = h + p * Hd;
        float* hn = h + (p ^ 1) * Hd;
        for (int i = tid; i < Hd; i += NTHR) hS[i] = hc[i];  // stage h in LDS
        __syncthreads();
        const float* gi = Gi + (size_t)t * G3;
        #pragma unroll
        for (int e = 0; e < 4; ++e) {
            const int j = blk * 32 + wv * 4 + e;
            const float* w0 = eWhh + (size_t)j * Hd;
            const float* w1 = eWhh + (size_t)(Hd + j) * Hd;
            const float* w2 = eWhh + (size_t)(2 * Hd + j) * Hd;
            float s0 = 0.f, s1 = 0.f, s2 = 0.f;
            for (int k = lane * 4; k < Hd; k += 128) {
                float4 hv = *(const float4*)(hS + k);
                float4 a0 = *(const float4*)(w0 + k);
                float4 a1 = *(const float4*)(w1 + k);
                float4 a2 = *(const float4*)(w2 + k);
                s0 += a0.x*hv.x + a0.y*hv.y + a0.z*hv.z + a0.w*hv.w;
                s1 += a1.x*hv.x + a1.y*hv.y + a1.z*hv.z + a1.w*hv.w;
                s2 += a2.x*hv.x + a2.y*hv.y + a2.z*hv.z + a2.w*hv.w;
            }
            s0 = wave_sum(s0); s1 = wave_sum(s1); s2 = wave_sum(s2);
            if (lane == 0) {
                float r  = sigm(gi[j]          + s0 + eBhh[j]);
                float z  = sigm(gi[Hd + j]     + s1 + eBhh[Hd + j]);
                float nn = tanhf(gi[2*Hd + j] + r * (s2 + eBhh[2*Hd + j]));
                hn[j] = (1.0f - z) * nn + z * hS[j];
            }
        }
        p ^= 1;
        grid_barrier(cnt, gen);
    }

    // ---------------- decoder: T autoregressive steps ----------------
    for (int t = 0; t < T; ++t) {
        const float* hc = h + p * Hd;
        float* hnw = h + (p ^ 1) * Hd;
        const int idx = atomicAdd(idxp, 0);      // -1 on first step
        for (int i = tid; i < Hd; i += NTHR) hS[i] = hc[i];
        __syncthreads();
        #pragma unroll
        for (int e = 0; e < 4; ++e) {
            const int j = blk * 32 + wv * 4 + e;
            const float* w0 = dWhh + (size_t)j * Hd;
            const float* w1 = dWhh + (size_t)(Hd + j) * Hd;
            const float* w2 = dWhh + (size_t)(2 * Hd + j) * Hd;
            float s0 = 0.f, s1 = 0.f, s2 = 0.f;
            for (int k = lane * 4; k < Hd; k += 128) {
                float4 hv = *(const float4*)(hS + k);
                float4 a0 = *(const float4*)(w0 + k);
                float4 a1 = *(const float4*)(w1 + k);
                float4 a2 = *(const float4*)(w2 + k);
                s0 += a0.x*hv.x + a0.y*hv.y + a0.z*hv.z + a0.w*hv.w;
                s1 += a1.x*hv.x + a1.y*hv.y + a1.z*hv.z + a1.w*hv.w;
                s2 += a2.x*hv.x + a2.y*hv.y + a2.z*hv.z + a2.w*hv.w;
            }
            s0 = wave_sum(s0); s1 = wave_sum(s1); s2 = wave_sum(s2);
            if (lane == 0) {
                // one-hot feedback -> gi is a column gather of dec_w_ih
                float gr = dBih[j], gz = dBih[Hd + j], gn = dBih[2*Hd + j];
                if (idx >= 0) {
                    gr += dWih[(size_t)j * Od + idx];
                    gz += dWih[(size_t)(Hd + j) * Od + idx];
                    gn += dWih[(size_t)(2*Hd + j) * Od + idx];
                }
                float r  = sigm(gr + s0 + dBhh[j]);
                float z  = sigm(gz + s1 + dBhh[Hd + j]);
                float nn = tanhf(gn + r * (s2 + dBhh[2*Hd + j]));
                hnw[j] = (1.0f - z) * nn + z * hS[j];
            }
        }
        p ^= 1;
        grid_barrier(cnt, gen);

        // logits: 512 dot-products distributed (8 per block, 1 per wave)
        {
            const float* hn2 = h + p * Hd;       // freshly written h
            const int o = blk * 8 + wv;
            const float* wr = h2oW + (size_t)o * Hd;
            float s = 0.f;
            for (int k = lane * 4; k < Hd; k += 128) {
                float4 hv = *(const float4*)(hn2 + k);
                float4 a  = *(const float4*)(wr + k);
                s += a.x*hv.x + a.y*hv.y + a.z*hv.z + a.w*hv.w;
            }
            s = wave_sum(s);
            if (lane == 0) logits[o] = s + h2oB[o];
        }
        grid_barrier(cnt, gen);

        // block 0: log_softmax + argmax + output write
        if (blk == 0) {
            float x0 = logits[tid], x1 = logits[tid + 256];
            red[tid]  = fmaxf(x0, x1);
            redi[tid] = (x1 > x0) ? (tid + 256) : tid;
            __syncthreads();
            for (int s2 = 128; s2 > 0; s2 >>= 1) {
                if (tid < s2 && red[tid + s2] > red[tid]) {
                    red[tid] = red[tid + s2]; redi[tid] = redi[tid + s2];
                }
                __syncthreads();
            }
            const float gm = red[0];
            const int   am = redi[0];
            __syncthreads();
            red[tid] = __expf(x0 - gm) + __expf(x1 - gm);
            __syncthreads();
            for (int s2 = 128; s2 > 0; s2 >>= 1) {
                if (tid < s2) red[tid] += red[tid + s2];
                __syncthreads();
            }
            const float lse = gm + __logf(red[0]);
            out[(size_t)t * Od + tid]       = x0 - lse;
            out[(size_t)t * Od + tid + 256] = x1 - lse;
            if (tid == 0) atomicExch(idxp, am);
        }
        grid_barrier(cnt, gen);
    }
}

// ---------------------------------------------------------------------------
extern "C" void kernel_launch(void* const* d_in, const int* in_sizes, int n_in,
                              void* d_out, int out_size, void* d_ws, size_t ws_size,
                              hipStream_t stream) {
    const float* input_ = (const float*)d_in[0];
    const float* eWih   = (const float*)d_in[1];
    const float* eWhh   = (const float*)d_in[2];
    const float* eBih   = (const float*)d_in[3];
    const float* eBhh   = (const float*)d_in[4];
    const float* dWih   = (const float*)d_in[5];
    const float* dWhh   = (const float*)d_in[6];
    const float* dBih   = (const float*)d_in[7];
    const float* dBhh   = (const float*)d_in[8];
    const float* h2oW   = (const float*)d_in[9];
    const float* h2oB   = (const float*)d_in[10];

    const int L = in_sizes[0] / Id;   // 4096
    const int T = out_size / Od;      // 30

    char* ws = (char*)d_ws;
    unsigned* cnt  = (unsigned*)(ws);
    unsigned* gen  = (unsigned*)(ws + 8);
    int*      idxp = (int*)(ws + 16);
    float*    h      = (float*)(ws + 64);                       // 2*2048 floats
    float*    logits = (float*)(ws + 64 + 2 * Hd * sizeof(float));
    float*    Gi     = (float*)(ws + 32768);                    // L*6144 floats

    gru_init_kernel<<<1, 256, 0, stream>>>(h, cnt, gen, idxp);

    const int waves = (L / 16) * (G3 / 64);      // one wave per 16x64 strip
    gi_gemm_kernel<<<(waves + 7) / 8, NTHR, 0, stream>>>(input_, eWih, eBih, Gi, L);

    rnn_kernel<<<NBLK, NTHR, 0, stream>>>(Gi, eWhh, eBhh, dWih, dWhh, dBih, dBhh,
                                          h2oW, h2oB, h, logits, cnt, gen, idxp,
                                          (float*)d_out, L, T);
}